// MarketEncoder_56556129354309
// MI455X (gfx1250) — compile-verified
//
#include <hip/hip_runtime.h>
#include <math.h>

typedef __attribute__((ext_vector_type(2))) float v2f;
typedef __attribute__((ext_vector_type(8))) float v8f;

#define SEQ     4096
#define IN_DIM  64
#define DM      1024
#define GDIM    4096        // 4*DM
#define REC_BLOCKS 64
#define REC_TPB    256
#define WPAD    1028        // padded LDS row stride: 1028 % 64 == 4 -> minimal b128 bank conflicts

__device__ __forceinline__ float sigmoidf(float x){ return 1.0f/(1.0f+expf(-x)); }
__device__ __forceinline__ float lrelu(float x){ return x > 0.0f ? x : 0.01f*x; }

// ---------- zero sync counters + h double buffers (deterministic per launch) ----------
__global__ void k_init(unsigned* __restrict__ p, int n){
  int i = blockIdx.x*blockDim.x + threadIdx.x;
  if (i < n) p[i] = 0u;
}

// ---------- x1 = leaky_relu(in @ W1^T + b1)  : M=4096 N=1024 K=64, fp32 WMMA ----------
__global__ void k_proj(const float* __restrict__ in, const float* __restrict__ W1,
                       const float* __restrict__ b1, float* __restrict__ x1){
  const int lane = threadIdx.x & 31;
  const int wave = threadIdx.x >> 5;
  const int tile = blockIdx.x * 8 + wave;   // 16384 tiles (256 x 64)
  const int tm = tile >> 6;
  const int tn = tile & 63;
  const int half = lane >> 4;
  const int idx  = lane & 15;

  const float* arow = in + (size_t)(tm*16 + idx)*IN_DIM;
  const float* brow = W1 + (size_t)(tn*16 + idx)*IN_DIM;

  v8f c = {};
  #pragma unroll
  for (int k0 = 0; k0 < IN_DIM; k0 += 4){
    v2f a = *(const v2f*)(arow + k0 + 2*half);
    v2f b = *(const v2f*)(brow + k0 + 2*half);
    c = __builtin_amdgcn_wmma_f32_16x16x4_f32(false, a, false, b, (short)0, c, false, false);
  }
  const float bias = b1[tn*16 + idx];
  #pragma unroll
  for (int v = 0; v < 8; ++v){
    int row = tm*16 + v + 8*half;
    x1[(size_t)row*DM + tn*16 + idx] = lrelu(c[v] + bias);
  }
}

// ---------- xg = x @ Wih^T + (bih+bhh) : M=4096 N=4096 K=1024, 2x2 tile blocking ----------
__global__ void k_gates(const float* __restrict__ x, const float* __restrict__ Wih,
                        const float* __restrict__ bih, const float* __restrict__ bhh,
                        float* __restrict__ xg){
  const int lane = threadIdx.x & 31;
  const int wave = threadIdx.x >> 5;
  const int tile = blockIdx.x * 8 + wave;   // 16384 wave-tiles of 32x32 (128 x 128)
  const int tm = tile >> 7;
  const int tn = tile & 127;
  const int half = lane >> 4;
  const int idx  = lane & 15;

  const float* a0row = x   + (size_t)(tm*32 + idx)*DM;
  const float* a1row = a0row + (size_t)16*DM;
  const float* b0row = Wih + (size_t)(tn*32 + idx)*DM;
  const float* b1row = b0row + (size_t)16*DM;

  v8f c00 = {}, c01 = {}, c10 = {}, c11 = {};
  for (int k0 = 0; k0 < DM; k0 += 4){
    const int ko = k0 + 2*half;
    v2f a0 = *(const v2f*)(a0row + ko);
    v2f a1 = *(const v2f*)(a1row + ko);
    v2f b0 = *(const v2f*)(b0row + ko);
    v2f b1 = *(const v2f*)(b1row + ko);
    c00 = __builtin_amdgcn_wmma_f32_16x16x4_f32(false, a0, false, b0, (short)0, c00, false, false);
    c01 = __builtin_amdgcn_wmma_f32_16x16x4_f32(false, a0, false, b1, (short)0, c01, false, false);
    c10 = __builtin_amdgcn_wmma_f32_16x16x4_f32(false, a1, false, b0, (short)0, c10, false, false);
    c11 = __builtin_amdgcn_wmma_f32_16x16x4_f32(false, a1, false, b1, (short)0, c11, false, false);
  }
  const int n0 = tn*32 + idx, n1 = n0 + 16;
  const float bb0 = bih[n0] + bhh[n0];
  const float bb1 = bih[n1] + bhh[n1];
  #pragma unroll
  for (int v = 0; v < 8; ++v){
    int r0 = tm*32 + v + 8*half;
    int r1 = r0 + 16;
    xg[(size_t)r0*GDIM + n0] = c00[v] + bb0;
    xg[(size_t)r0*GDIM + n1] = c01[v] + bb1;
    xg[(size_t)r1*GDIM + n0] = c10[v] + bb0;
    xg[(size_t)r1*GDIM + n1] = c11[v] + bb1;
  }
}

// ---------- sequential LSTM recurrence: 64 persistent WGs, device-wide barrier/step ----------
// CDNA5: 320KB LDS per WGP -> each WG caches its entire 64-row (256KB) Whh slice in LDS.
__global__ void k_recur(const float* __restrict__ xg, const float* __restrict__ Whh,
                        float* __restrict__ hseq, float* __restrict__ hbuf,
                        volatile unsigned* ctr){
  __shared__ float wsh[64*WPAD];   // 263,168 B weight slice (padded rows)
  __shared__ float hsh[DM];
  __shared__ float part[REC_TPB];
  __shared__ float gsh[64];
  __shared__ float csh[16];
  const int tid = threadIdx.x;
  const int r    = tid & 63;     // row within this block's 64 gate rows
  const int q    = tid >> 6;     // K quarter (0..3)
  const int gate = r >> 4;
  const int u    = r & 15;
  const int grow = gate*DM + blockIdx.x*16 + u;      // global gate row (i,f,g,o order)

  if (tid < 16) csh[tid] = 0.0f;

  // one-time preload of this WG's Whh slice into LDS
  {
    const float* src = Whh + (size_t)grow*DM + q*256;
    float* dst = wsh + r*WPAD + q*256;
    #pragma unroll 4
    for (int k = 0; k < 256; k += 4)
      *(float4*)(dst + k) = *(const float4*)(src + k);
  }
  __syncthreads();

  const float* wl = wsh + r*WPAD + q*256;  // 16B-aligned (1028*4 = 257*16)

  for (int t = 0; t < SEQ; ++t){
    // stage h(t-1) into LDS (double-buffered in global)
    const float* hb = hbuf + (size_t)(t & 1)*DM;
    *(float4*)(hsh + tid*4) = *(const float4*)(hb + tid*4);
    __syncthreads();

    // partial dot: Whh[grow, q*256 : q*256+256] . h  (weights + h both in LDS)
    float s = 0.0f;
    const float* hq = hsh + q*256;
    #pragma unroll 8
    for (int k = 0; k < 256; k += 4){
      float4 w = *(const float4*)(wl + k);
      s += w.x*hq[k] + w.y*hq[k+1] + w.z*hq[k+2] + w.w*hq[k+3];
    }
    part[tid] = s;
    __syncthreads();

    if (tid < 64){
      int gr = (tid >> 4)*DM + blockIdx.x*16 + (tid & 15);
      gsh[tid] = part[tid] + part[tid+64] + part[tid+128] + part[tid+192]
               + xg[(size_t)t*GDIM + gr];
    }
    __syncthreads();

    if (tid < 16){
      float gi = gsh[tid], gf = gsh[16+tid], gg = gsh[32+tid], go = gsh[48+tid];
      float cnew = sigmoidf(gf)*csh[tid] + sigmoidf(gi)*tanhf(gg);
      csh[tid] = cnew;
      float hnew = sigmoidf(go)*tanhf(cnew);
      int d = blockIdx.x*16 + tid;
      hbuf[(size_t)((t+1) & 1)*DM + d] = hnew;
      hseq[(size_t)t*DM + d] = hnew;
    }
    __threadfence();
    __syncthreads();
    if (tid == 0){
      atomicAdd((unsigned*)(ctr + t), 1u);
      while (ctr[t] < (unsigned)REC_BLOCKS) { }
    }
    __syncthreads();
    __threadfence();
  }
}

// ---------- out = leaky_relu(h_T) ----------
__global__ void k_final(const float* __restrict__ hlast, float* __restrict__ out){
  int i = blockIdx.x*blockDim.x + threadIdx.x;
  if (i < DM) out[i] = lrelu(hlast[i]);
}

extern "C" void kernel_launch(void* const* d_in, const int* in_sizes, int n_in,
                              void* d_out, int out_size, void* d_ws, size_t ws_size,
                              hipStream_t stream){
  const float* in  = (const float*)d_in[0];
  const float* W1  = (const float*)d_in[1];
  const float* b1  = (const float*)d_in[2];
  const float* Wih = (const float*)d_in[3];
  const float* Whh = (const float*)d_in[4];
  const float* bih = (const float*)d_in[5];
  const float* bhh = (const float*)d_in[6];
  float* out = (float*)d_out;

  char* ws = (char*)d_ws;
  float*    xA   = (float*)(ws);                       // 16MB: x1, later reused as hseq2
  float*    xg   = (float*)(ws + (size_t)(16u<<20));   // 64MB gate preactivations
  float*    hs1  = (float*)(ws + (size_t)(80u<<20));   // 16MB layer-1 h sequence
  unsigned* ctr1 = (unsigned*)(ws + (size_t)(96u<<20));
  unsigned* ctr2 = ctr1 + SEQ;
  float*    hbuf1 = (float*)(ctr2 + SEQ);
  float*    hbuf2 = hbuf1 + 2*DM;

  // zero counters (2*4096 u32) + h double-buffers (2*2048 f32) = 12288 words
  k_init<<<48, 256, 0, stream>>>(ctr1, SEQ*2 + 4*DM);

  k_proj<<<2048, 256, 0, stream>>>(in, W1, b1, xA);

  // layer 0
  k_gates<<<2048, 256, 0, stream>>>(xA, Wih, bih, bhh, xg);
  k_recur<<<REC_BLOCKS, REC_TPB, 0, stream>>>(xg, Whh, hs1, hbuf1, ctr1);

  // layer 1
  k_gates<<<2048, 256, 0, stream>>>(hs1, Wih + (size_t)GDIM*DM, bih + GDIM, bhh + GDIM, xg);
  k_recur<<<REC_BLOCKS, REC_TPB, 0, stream>>>(xg, Whh + (size_t)GDIM*DM, xA, hbuf2, ctr2);

  k_final<<<4, 256, 0, stream>>>(xA + (size_t)(SEQ-1)*DM, out);
}